// _RegridFromLatLon_38070590111802
// MI455X (gfx1250) — compile-verified
//
#include <hip/hip_runtime.h>

// Bilinear regrid from periodic lat-lon grid onto unstructured points.
// x: [B=8, NLAT=721, NLON=1440] f32; xi: [N,2] (lon,lat); out: [N,B] f32.
//
// MI455X strategy:
//  - grid x (33 MB) is L2-resident (192 MB L2): regular-temporal gathers
//  - xi / out are read-once / write-once streams: nontemporal hints
//  - lon/lat coordinate arrays staged into LDS once per block
//  - analytic uniform-grid index guess + exact searchsorted fix-up
//  - global_prefetch_b8 of both grid rows issued before the weight divides,
//    hiding L2 gather latency behind the transcendental-latency math

typedef float v2f __attribute__((ext_vector_type(2)));
typedef float v4f __attribute__((ext_vector_type(4)));

#define RG_NLON 1440
#define RG_NLAT 721

__global__ __launch_bounds__(256) void regrid_latlon_kernel(
    const float* __restrict__ x,        // [B, NLAT, NLON]
    const float* __restrict__ lon_src,  // [NLON]
    const float* __restrict__ lat_src,  // [NLAT]
    const float* __restrict__ xi,       // [N, 2]
    float* __restrict__ out,            // [N, B]
    int N)
{
    __shared__ float s_lon[RG_NLON];
    __shared__ float s_lat[RG_NLAT];

    // Stage small coordinate arrays into LDS (cooperative, coalesced b32s).
    for (int t = threadIdx.x; t < RG_NLON; t += blockDim.x) s_lon[t] = lon_src[t];
    for (int t = threadIdx.x; t < RG_NLAT; t += blockDim.x) s_lat[t] = lat_src[t];
    __syncthreads();

    const int n = blockIdx.x * blockDim.x + threadIdx.x;
    if (n >= N) return;

    // Streaming (read-once) query load: nontemporal.
    const v2f q = __builtin_nontemporal_load(((const v2f*)xi) + n);
    const float lon = q.x;
    const float lat = q.y;

    // --- longitude cell: uniform 0.25-degree guess, then exact fix-up ---
    int i = (int)(lon * 4.0f);
    i = i < 0 ? 0 : (i > RG_NLON - 1 ? RG_NLON - 1 : i);
    while (i > 0 && lon < s_lon[i]) --i;
    while (i < RG_NLON - 1 && lon >= s_lon[i + 1]) ++i;

    // --- latitude cell: uniform guess, then exact fix-up (clip to nlat-2) ---
    int j = (int)((lat + 90.0f) * 4.0f);
    j = j < 0 ? 0 : (j > RG_NLAT - 2 ? RG_NLAT - 2 : j);
    while (j > 0 && lat < s_lat[j]) --j;
    while (j < RG_NLAT - 2 && lat >= s_lat[j + 1]) ++j;

    const long long rowoff  = (long long)j * RG_NLON + i;
    const long long bstride = (long long)RG_NLAT * RG_NLON;

    // Kick the gather cachelines toward L0 while we do the weight divides.
    // global_prefetch_b8 is side-effect-free (no VGPR return, no counters).
    #pragma unroll
    for (int b = 0; b < 8; ++b) {
        const float* p = x + (long long)b * bstride + rowoff;
        __builtin_prefetch(p, 0, 3);            // row j   line
        __builtin_prefetch(p + RG_NLON, 0, 3);  // row j+1 line
    }

    // Weights from the actual coordinate arrays (faithful to reference).
    const float lon_i  = s_lon[i];
    const float lon_i1 = (i == RG_NLON - 1) ? 360.0f : s_lon[i + 1];
    const float wlon   = (lon - lon_i) / (lon_i1 - lon_i);

    const float lat_j  = s_lat[j];
    const float lat_j1 = s_lat[j + 1];
    const float wlat   = (lat - lat_j) / (lat_j1 - lat_j);

    // Wrapped column offset: column i+1 is column 0 when i == NLON-1.
    const int dcol = (i == RG_NLON - 1) ? (1 - RG_NLON) : 1;

    float res[8];
    #pragma unroll
    for (int b = 0; b < 8; ++b) {
        // Grid gathers: regular-temporal so the 33 MB grid stays L2-resident.
        const float* p = x + (long long)b * bstride + rowoff;
        const float v00 = p[0];
        const float v10 = p[dcol];
        const float v01 = p[RG_NLON];
        const float v11 = p[RG_NLON + dcol];
        // Exact expression tree of the reference.
        res[b] = (1.0f - wlat) * ((1.0f - wlon) * v00 + wlon * v10)
               +         wlat  * ((1.0f - wlon) * v01 + wlon * v11);
    }

    // Write-once output: nontemporal 2x16B stores (wave writes contiguous 1KB).
    v4f lo = { res[0], res[1], res[2], res[3] };
    v4f hi = { res[4], res[5], res[6], res[7] };
    v4f* op = (v4f*)(out + (long long)n * 8);
    __builtin_nontemporal_store(lo, op);
    __builtin_nontemporal_store(hi, op + 1);
}

extern "C" void kernel_launch(void* const* d_in, const int* in_sizes, int n_in,
                              void* d_out, int out_size, void* d_ws, size_t ws_size,
                              hipStream_t stream) {
    const float* x       = (const float*)d_in[0];  // [B, 721, 1440]
    const float* lon_src = (const float*)d_in[1];  // [1440]
    const float* lat_src = (const float*)d_in[2];  // [721]
    const float* xi      = (const float*)d_in[3];  // [N, 2]
    float* out = (float*)d_out;                    // [N, B]

    const int N = in_sizes[3] / 2;

    (void)d_ws; (void)ws_size; (void)out_size; (void)n_in;

    const int block = 256;
    const int grid  = (N + block - 1) / block;
    regrid_latlon_kernel<<<grid, block, 0, stream>>>(x, lon_src, lat_src, xi, out, N);
}